// SirenImplicitGAN_35905926594974
// MI455X (gfx1250) — compile-verified
//
#include <hip/hip_runtime.h>

#define NITEMS 100000
#define LAT    256
#define BATCH  512
#define KSEL   101      // MANIFOLD_DIM + 1
#define NNCNT  100
#define COLT   (NITEMS / 16)   // 6250 item tiles
#define ROWT   (BATCH / 16)    // 32 batch tiles
#define NBLK   4               // N register blocking in dist kernel
#define CGRP   ((COLT + NBLK - 1) / NBLK)  // 1563 column groups
#define GT     7               // 7*16 = 112 >= 101 extended rows
#define TILES_PER_B (GT * GT)  // 49
#define EQCAP  160

typedef __attribute__((ext_vector_type(2))) float v2f;
typedef __attribute__((ext_vector_type(8))) float v8f;

// ---------------- kernel 1: per-item squared norms ----------------
__global__ __launch_bounds__(256) void item_norms_kernel(
    const float* __restrict__ latents, float* __restrict__ ln2) {
  int i = blockIdx.x * blockDim.x + threadIdx.x;
  if (i >= NITEMS) return;
  const float4* row = (const float4*)(latents + (long)i * LAT);
  float acc = 0.f;
#pragma unroll 4
  for (int j = 0; j < LAT / 4; ++j) {
    float4 v = row[j];
    acc += v.x * v.x + v.y * v.y + v.z * v.z + v.w * v.w;
  }
  ln2[i] = acc;
}

// ---------------- kernel 2: distance GEMM via V_WMMA_F32_16X16X4_F32 ----
// One wave computes a 16(batch) x 64(items) slab: 4 WMMA accumulators share
// one A fragment per K-step (1.25 b64 loads per WMMA instead of 2).
__global__ __launch_bounds__(256) void dist_wmma_kernel(
    const float* __restrict__ latents, const int* __restrict__ idx,
    const float* __restrict__ ln2, float* __restrict__ dist) {
  const int wave = threadIdx.x >> 5;
  const int lane = threadIdx.x & 31;
  const long t = (long)blockIdx.x * 8 + wave;   // 6252 blocks * 8 = 32*1563 waves
  const int rt = (int)(t / CGRP);
  const int cg = (int)(t % CGRP);
  const int ct0 = cg * NBLK;
  const int m  = lane & 15;
  const int kh = (lane >> 4) << 1;              // K sub-offset 0 or 2 (ISA layout)

  const float* __restrict__ Ap = latents + (long)idx[rt * 16 + m] * LAT;
  const float* Bp[NBLK];
#pragma unroll
  for (int j = 0; j < NBLK; ++j) {
    int item = (ct0 + j) * 16 + m;
    if (item >= NITEMS) item = NITEMS - 1;      // wave-uniform clamp, store skipped
    Bp[j] = latents + (long)item * LAT;
  }

  __builtin_prefetch(Bp[0], 0, 1);              // global_prefetch_b8

  v8f c[NBLK];
#pragma unroll
  for (int j = 0; j < NBLK; ++j) c[j] = (v8f){};

#pragma unroll 4
  for (int k0 = 0; k0 < LAT; k0 += 4) {
    const v2f a = *(const v2f*)(Ap + k0 + kh);
#pragma unroll
    for (int j = 0; j < NBLK; ++j) {
      const v2f b = *(const v2f*)(Bp[j] + k0 + kh);
      c[j] = __builtin_amdgcn_wmma_f32_16x16x4_f32(false, a, false, b,
                                                   (short)0, c[j], false, false);
    }
  }

  // C layout: VGPR r -> M = r + 8*(lane>>4), N = lane&15
  const int mbase = rt * 16 + ((lane >> 4) << 3);
  float la[8];
#pragma unroll
  for (int r = 0; r < 8; ++r) la[r] = ln2[idx[mbase + r]];

#pragma unroll
  for (int j = 0; j < NBLK; ++j) {
    const int ncol = (ct0 + j) * 16 + (lane & 15);
    if (ncol < NITEMS) {
      const float lb = ln2[ncol];
#pragma unroll
      for (int r = 0; r < 8; ++r) {
        float d2 = la[r] + lb - 2.0f * c[j][r];
        d2 = fmaxf(d2, 0.0f);                   // clip(d2, 0) as in reference
        dist[(long)(mbase + r) * NITEMS + ncol] = d2;
      }
    }
  }
}

// ---------------- kernel 3: exact top-101 (both directions) per row ------
// Radix select on the monotone uint image of non-negative floats; ties go to
// the lowest index (matches jax top_k); final 101 entries rank-sorted.
__global__ __launch_bounds__(256) void topk_kernel(
    const float* __restrict__ dist, int* __restrict__ nn, int* __restrict__ fari) {
  const int b = blockIdx.x;
  const int tid = threadIdx.x;
  const float* __restrict__ drow = dist + (long)b * NITEMS;

  __shared__ int hist[256];
  __shared__ unsigned int sPrefix;
  __shared__ int sWant;
  __shared__ int cLess, cEq;
  __shared__ unsigned int sKey[KSEL];
  __shared__ int sIdx[KSEL];
  __shared__ int eqIdx[EQCAP];
  __shared__ int outIdx[KSEL];

  for (int pass = 0; pass < 2; ++pass) {        // 0: nearest, 1: farthest
    unsigned int prefix = 0;
    int want = KSEL;
    for (int shift = 24; shift >= 0; shift -= 8) {
      hist[tid] = 0;
      __syncthreads();
      const unsigned int hiMask = (shift == 24) ? 0u : (0xFFFFFFFFu << (shift + 8));
      for (int i = tid; i < NITEMS; i += 256) {
        unsigned int key = __float_as_uint(drow[i]);
        if (pass) key = ~key;
        if ((key & hiMask) == prefix)
          atomicAdd(&hist[(key >> shift) & 255], 1);
      }
      __syncthreads();
      if (tid == 0) {
        int acc = 0;
        for (int bin = 0; bin < 256; ++bin) {
          int h = hist[bin];
          if (acc + h >= want) {
            sPrefix = prefix | ((unsigned int)bin << shift);
            sWant = want - acc;
            break;
          }
          acc += h;
        }
      }
      __syncthreads();
      prefix = sPrefix;
      want = sWant;
      __syncthreads();
    }
    const unsigned int T = prefix;              // exact 101st-smallest key
    if (tid == 0) { cLess = 0; cEq = 0; }
    __syncthreads();
    for (int i = tid; i < NITEMS; i += 256) {
      unsigned int key = __float_as_uint(drow[i]);
      if (pass) key = ~key;
      if (key < T) {
        int p = atomicAdd(&cLess, 1);
        if (p < KSEL) { sKey[p] = key; sIdx[p] = i; }
      } else if (key == T) {
        int p = atomicAdd(&cEq, 1);
        if (p < EQCAP) eqIdx[p] = i;
      }
    }
    __syncthreads();
    if (tid == 0) {
      int nl = cLess < KSEL ? cLess : KSEL;
      int ce = cEq < EQCAP ? cEq : EQCAP;
      for (int a = 1; a < ce; ++a) {            // ties: smallest indices first
        int v = eqIdx[a]; int j = a - 1;
        while (j >= 0 && eqIdx[j] > v) { eqIdx[j + 1] = eqIdx[j]; --j; }
        eqIdx[j + 1] = v;
      }
      int need = KSEL - nl;
      if (need > ce) need = ce;
      for (int e = 0; e < need; ++e) { sKey[nl + e] = T; sIdx[nl + e] = eqIdx[e]; }
      for (int p = nl + need; p < KSEL; ++p) { sKey[p] = 0xFFFFFFFFu; sIdx[p] = NITEMS + p; }
    }
    __syncthreads();
    if (tid < KSEL) {                           // rank sort: key asc, idx asc
      const unsigned int kj = sKey[tid];
      const int ij = sIdx[tid];
      int rank = 0;
      for (int l = 0; l < KSEL; ++l) {
        const unsigned int kl = sKey[l];
        const int il = sIdx[l];
        rank += (kl < kj || (kl == kj && il < ij)) ? 1 : 0;
      }
      outIdx[rank] = ij;
    }
    __syncthreads();
    int* dst = pass ? (fari + (long)b * KSEL) : (nn + (long)b * KSEL);
    if (tid < KSEL) dst[tid] = outIdx[tid];
    __syncthreads();
  }
}

// ---------------- kernel 4: gram + rhs via WMMA --------------------------
// Extended matrix E: rows 0..99 = sel (nn[1..100]), row 100 = z.
// E*E^T gives gram (0..99 x 0..99) and rhs (col 100).
__global__ __launch_bounds__(256) void gram_wmma_kernel(
    const float* __restrict__ latents, const int* __restrict__ idx,
    const int* __restrict__ nn, float* __restrict__ gram, float* __restrict__ rhs) {
  const int wave = threadIdx.x >> 5;
  const int lane = threadIdx.x & 31;
  const long t = (long)blockIdx.x * 8 + wave;   // 3136 blocks * 8 = 512*49 tiles
  const int b  = (int)(t / TILES_PER_B);
  const int tt = (int)(t % TILES_PER_B);
  const int ti = tt / GT, tj = tt % GT;
  const int m  = lane & 15;
  const int kh = (lane >> 4) << 1;

  const int* __restrict__ nb = nn + (long)b * KSEL;
  auto erow = [&](int r) -> int {
    if (r < NNCNT) return nb[1 + r];
    if (r == NNCNT) return idx[b];
    return nb[1];                                // dummy rows; never stored
  };
  const float* __restrict__ Ap = latents + (long)erow(ti * 16 + m) * LAT;
  const float* __restrict__ Bp = latents + (long)erow(tj * 16 + m) * LAT;

  v8f c = {};
#pragma unroll 8
  for (int k0 = 0; k0 < LAT; k0 += 4) {
    v2f a  = *(const v2f*)(Ap + k0 + kh);
    v2f bb = *(const v2f*)(Bp + k0 + kh);
    c = __builtin_amdgcn_wmma_f32_16x16x4_f32(false, a, false, bb,
                                              (short)0, c, false, false);
  }
  const int gj = tj * 16 + (lane & 15);
  const int mbase = ti * 16 + ((lane >> 4) << 3);
#pragma unroll
  for (int r = 0; r < 8; ++r) {
    const int gi = mbase + r;
    if (gi < NNCNT && gj < NNCNT)
      gram[((long)b * NNCNT + gi) * NNCNT + gj] = c[r];
    else if (gi < NNCNT && gj == NNCNT)
      rhs[(long)b * NNCNT + gi] = c[r];
  }
}

// ---------------- kernel 5: 100x100 solve in LDS + all outputs ----------
__global__ __launch_bounds__(256) void solve_out_kernel(
    const float* __restrict__ latents, const int* __restrict__ idx,
    const int* __restrict__ nn, const int* __restrict__ fari,
    const float* __restrict__ gram, const float* __restrict__ rhs,
    float* __restrict__ out_lin, float* __restrict__ out_dist,
    float* __restrict__ out_far) {
  const int b = blockIdx.x;
  const int tid = threadIdx.x;
  __shared__ __align__(16) float sA[NNCNT * NNCNT];  // 40 KB (320KB-LDS WGP)
  __shared__ float sr[NNCNT];
  __shared__ float sw[NNCNT];
  __shared__ float sz[LAT];

  // Stage gram via async tensor path: memory -> LDS, no VGPR bounce.
  // Generic LDS pointers on AMDGPU carry the LDS byte offset in bits [31:0].
  {
    const float* __restrict__ gsrc = gram + (long)b * NNCNT * NNCNT;
    for (int e = tid; e < (NNCNT * NNCNT) / 4; e += 256) {      // 2500 x B128
      unsigned lds_off = (unsigned)(unsigned long long)(const void*)&sA[e * 4];
      unsigned long long ga = (unsigned long long)(gsrc + e * 4);
      asm volatile("global_load_async_to_lds_b128 %0, %1, off"
                   :: "v"(lds_off), "v"(ga) : "memory");
    }
    asm volatile("s_wait_asynccnt 0" ::: "memory");
  }
  if (tid < NNCNT) sr[tid] = rhs[(long)b * NNCNT + tid];
  const float* __restrict__ zp = latents + (long)idx[b] * LAT;
  for (int i = tid; i < LAT; i += 256) sz[i] = zp[i];
  __syncthreads();

  // forward elimination (gram is SPD; no pivoting)
  for (int k = 0; k < NNCNT - 1; ++k) {
    const float piv = sA[k * NNCNT + k];
    for (int i = k + 1 + tid; i < NNCNT; i += 256) {
      const float f = sA[i * NNCNT + k] / piv;
      for (int j = k; j < NNCNT; ++j) sA[i * NNCNT + j] -= f * sA[k * NNCNT + j];
      sr[i] -= f * sr[k];
    }
    __syncthreads();
  }
  if (tid == 0) {                               // back substitution
    for (int k = NNCNT - 1; k >= 0; --k) {
      float acc = sr[k];
      for (int j = k + 1; j < NNCNT; ++j) acc -= sA[k * NNCNT + j] * sw[j];
      sw[k] = acc / sA[k * NNCNT + k];
    }
  }
  __syncthreads();

  const int* __restrict__ nb = nn + (long)b * KSEL;
  for (int d = tid; d < LAT; d += 256) {        // latents_linear
    float acc = 0.f;
    for (int k = 0; k < NNCNT; ++k)
      acc += sw[k] * latents[(long)nb[1 + k] * LAT + d];
    out_lin[(long)b * LAT + d] = acc;
  }
  if (tid < 20) {                               // latents_dist (20 nearest)
    const float* sp = latents + (long)nb[1 + tid] * LAT;
    float acc = 0.f;
    for (int d = 0; d < LAT; ++d) { float df = sp[d] - sz[d]; acc += df * df; }
    out_dist[(long)b * 20 + tid] = sqrtf(acc);
  }
  for (int j = tid; j < KSEL; j += 256) {       // latents_dist_large (101 far)
    const float* sp = latents + (long)fari[(long)b * KSEL + j] * LAT;
    float acc = 0.f;
    for (int d = 0; d < LAT; ++d) { float df = sp[d] - sz[d]; acc += df * df; }
    out_far[(long)b * KSEL + j] = sqrtf(acc);
  }
}

extern "C" void kernel_launch(void* const* d_in, const int* in_sizes, int n_in,
                              void* d_out, int out_size, void* d_ws, size_t ws_size,
                              hipStream_t stream) {
  const float* latents = (const float*)d_in[0];
  const int* idx = (const int*)d_in[1];

  // workspace layout (~226 MB): ln2 | dist | nn | far | gram | rhs
  float* ln2  = (float*)d_ws;
  float* dist = ln2 + NITEMS;
  int*   nn   = (int*)(dist + (long)BATCH * NITEMS);
  int*   fari = nn + BATCH * KSEL;
  float* gram = (float*)(fari + BATCH * KSEL);
  float* rhs  = gram + (long)BATCH * NNCNT * NNCNT;

  float* out_lin  = (float*)d_out;                      // 512 x 256
  float* out_dist = out_lin + (long)BATCH * LAT;        // 512 x 20
  float* out_far  = out_dist + (long)BATCH * 20;        // 512 x 101

  hipLaunchKernelGGL(item_norms_kernel, dim3((NITEMS + 255) / 256), dim3(256), 0,
                     stream, latents, ln2);
  hipLaunchKernelGGL(dist_wmma_kernel, dim3(ROWT * CGRP / 8), dim3(256), 0,
                     stream, latents, idx, ln2, dist);
  hipLaunchKernelGGL(topk_kernel, dim3(BATCH), dim3(256), 0, stream, dist, nn, fari);
  hipLaunchKernelGGL(gram_wmma_kernel, dim3(BATCH * TILES_PER_B / 8), dim3(256), 0,
                     stream, latents, idx, nn, gram, rhs);
  hipLaunchKernelGGL(solve_out_kernel, dim3(BATCH), dim3(256), 0, stream,
                     latents, idx, nn, fari, gram, rhs, out_lin, out_dist, out_far);
}